// SwinTransformerBlock_kv_64536178589753
// MI455X (gfx1250) — compile-verified
//
#include <hip/hip_runtime.h>

// ---------------------------------------------------------------------------
// Deformable 3D window attention (Swin-style) for MI455X / gfx1250.
// fp32 end-to-end; all GEMMs routed through V_WMMA_F32_16X16X4_F32.
// Conv GEMMs register-block 4 M-tiles per wave to raise WMMA:VMEM ratio.
// ---------------------------------------------------------------------------

typedef float v2f __attribute__((ext_vector_type(2)));
typedef float v8f __attribute__((ext_vector_type(8)));

#define BB   2
#define DIM  192
#define NH   6
#define HD   32
#define SS   16
#define HH   32
#define WW   32
#define SHW  (SS*HH*WW)   // 16384
#define NW   256          // windows per batch (4*8*8)
#define NWIN 64           // tokens per window
#define SCALE_Q 0.17677669529663687f  // 32^-0.5

__device__ __forceinline__ v8f wmma_f32(v2f a, v2f b, v8f c) {
  // D = A(16x4,f32) * B(4x16,f32) + C(16x16,f32)
  return __builtin_amdgcn_wmma_f32_16x16x4_f32(false, a, false, b, (short)0, c,
                                               false, false);
}

// ---------------------------------------------------------------------------
// Kernel 1: window avg-pool -> leaky relu -> offset / scale 1x1 convs.
// One block per (b, window); 192 threads (one per channel).
// ---------------------------------------------------------------------------
__global__ void k_pool_offsets(const float* __restrict__ x,
                               const float* __restrict__ w_off,
                               const float* __restrict__ b_off,
                               const float* __restrict__ w_sc,
                               const float* __restrict__ b_sc,
                               float* __restrict__ off_out,
                               float* __restrict__ sc_out) {
  __shared__ float pooled[DIM];
  int bw  = blockIdx.x;            // b*256 + win
  int b   = bw >> 8;
  int win = bw & 255;
  int wsz = win >> 6;              // window idx along S (0..3)
  int wyh = (win >> 3) & 7;        // along H (0..7)
  int wxw = win & 7;               // along W (0..7)
  int c = threadIdx.x;             // 0..191

  const float* xp = x + ((size_t)(b * DIM + c)) * SHW
                      + (wsz * 4) * 1024 + (wyh * 4) * 32 + (wxw * 4);
  float s = 0.f;
  #pragma unroll
  for (int i = 0; i < 4; ++i)
    #pragma unroll
    for (int j = 0; j < 4; ++j)
      #pragma unroll
      for (int l = 0; l < 4; ++l)
        s += xp[i * 1024 + j * 32 + l];
  s *= (1.0f / 64.0f);
  pooled[c] = (s >= 0.f) ? s : 0.01f * s;   // leaky relu
  __syncthreads();

  if (c < 36) {
    bool is_sc = (c >= 18);
    int  o     = is_sc ? (c - 18) : c;      // 0..17 = head*3 + ch
    const float* wrow = (is_sc ? w_sc : w_off) + o * DIM;
    float acc = is_sc ? b_sc[o] : b_off[o];
    for (int k = 0; k < DIM; ++k) acc += wrow[k] * pooled[k];
    int head = o / 3, ch = o % 3;
    size_t idx = (((size_t)b * NH + head) * 3 + ch) * NW + win;
    if (!is_sc) {
      const float div3[3] = {8.f, 8.f, 4.f};  // [W/ws, H/ws, S/ws]
      off_out[idx] = acc / div3[ch];
    } else {
      sc_out[idx] = acc;
    }
  }
}

// ---------------------------------------------------------------------------
// Kernel 2/4: 1x1x1 conv as GEMM  out[b, M, p] = W(MxK) * x[b, K, p] + bias.
// Each wave computes a 64(M) x 16(N) strip: 4 register-blocked 16x16 tiles
// sharing one B fragment per K-step (4 WMMA per B load-pair).
// A-frag layout (16x4 f32): lane&15 = M row, (lane>>4)*2 = K half.
// ---------------------------------------------------------------------------
template <int MROWS>
__global__ void __launch_bounds__(256)
k_conv1x1_wmma(const float* __restrict__ xin,
               const float* __restrict__ wgt,
               const float* __restrict__ bias,
               float* __restrict__ out) {
  const int MG = MROWS / 64;      // M-tile groups of 4 (9 or 3)
  const int NT = SHW / 16;        // 1024
  int lane = threadIdx.x & 31;
  int wave = threadIdx.x >> 5;
  int tile = blockIdx.x * 8 + wave;     // over B*MG*NT
  int b    = tile / (MG * NT);
  int rem  = tile % (MG * NT);
  int mg   = rem / NT;
  int nt   = rem % NT;

  int mrow  = lane & 15;           // A row / B column
  int khalf = (lane >> 4) * 2;     // K sub-offset per lane group

  const float* A0 = wgt + (size_t)(mg * 64 + mrow) * DIM + khalf;
  const float* Bx = xin + (size_t)b * DIM * SHW + nt * 16 + mrow;

  v8f c0 = {}, c1 = {}, c2 = {}, c3 = {};
  for (int k0 = 0; k0 < DIM; k0 += 4) {
    v2f bb;
    bb.x = Bx[(size_t)(k0 + khalf + 0) * SHW];
    bb.y = Bx[(size_t)(k0 + khalf + 1) * SHW];
    // prefetch next K-step's x rows into cache (global_prefetch_b8)
    int kp = (k0 + 4 < DIM) ? (k0 + 4) : k0;
    __builtin_prefetch(&Bx[(size_t)(kp + khalf) * SHW], 0, 3);
    v2f a0, a1, a2, a3;
    a0.x = A0[k0 + 0 * 16 * DIM]; a0.y = A0[k0 + 0 * 16 * DIM + 1];
    a1.x = A0[k0 + 1 * 16 * DIM]; a1.y = A0[k0 + 1 * 16 * DIM + 1];
    a2.x = A0[k0 + 2 * 16 * DIM]; a2.y = A0[k0 + 2 * 16 * DIM + 1];
    a3.x = A0[k0 + 3 * 16 * DIM]; a3.y = A0[k0 + 3 * 16 * DIM + 1];
    c0 = wmma_f32(a0, bb, c0);
    c1 = wmma_f32(a1, bb, c1);
    c2 = wmma_f32(a2, bb, c2);
    c3 = wmma_f32(a3, bb, c3);
  }

  // D layout: VGPR r -> M = (lane>=16 ? 8 : 0) + r, N = lane&15
  int n     = nt * 16 + (lane & 15);
  int mbase = mg * 64 + (lane >> 4) * 8;
  float* op = out + (size_t)b * MROWS * SHW + n;
  #pragma unroll
  for (int r = 0; r < 8; ++r) {
    op[(size_t)(mbase +  0 + r) * SHW] = c0[r] + bias[mbase +  0 + r];
    op[(size_t)(mbase + 16 + r) * SHW] = c1[r] + bias[mbase + 16 + r];
    op[(size_t)(mbase + 32 + r) * SHW] = c2[r] + bias[mbase + 32 + r];
    op[(size_t)(mbase + 48 + r) * SHW] = c3[r] + bias[mbase + 48 + r];
  }
}

// ---------------------------------------------------------------------------
// Kernel 3: per (b, head, window) deformable window attention.
// 256 threads = 8 wave32. Trilinear-sample K/V into LDS, QK^T & AV via WMMA.
// All WMMA regions have constant trip counts (no EXEC masking around WMMA).
// ---------------------------------------------------------------------------
__global__ void __launch_bounds__(256)
k_attn(const float* __restrict__ qkv,
       const float* __restrict__ off_arr,
       const float* __restrict__ sc_arr,
       const float* __restrict__ rpb_table,
       float* __restrict__ outp) {
  __shared__ float qw[NWIN][HD];
  __shared__ float kw[NWIN][HD];
  __shared__ float vw[NWIN][HD];
  __shared__ float attn[NWIN][NWIN];
  __shared__ int   gix[NWIN][3];
  __shared__ float gfr[NWIN][3];

  int blk  = blockIdx.x;           // (b*NH + head)*NW + win
  int win  = blk % NW;
  int bh   = blk / NW;
  int head = bh % NH;
  int b    = bh / NH;
  int wsz  = win >> 6, wyh = (win >> 3) & 7, wxw = win & 7;
  int tid  = threadIdx.x;

  // --- per-token deformed sampling coordinates (identity grid + rel*sc + off)
  if (tid < NWIN) {
    int i = tid >> 4, j = (tid >> 2) & 3, l = tid & 3;   // (s,h,w) in window
    int zi = wsz * 4 + i, yi = wyh * 4 + j, xi = wxw * 4 + l;
    size_t pbase = ((size_t)bh * 3) * NW + win;
    float ax = sc_arr[pbase + 0 * NW], ay = sc_arr[pbase + 1 * NW], az = sc_arr[pbase + 2 * NW];
    float ox = off_arr[pbase + 0 * NW], oy = off_arr[pbase + 1 * NW], oz = off_arr[pbase + 2 * NW];
    float relx = (l - 1.5f) * (2.0f / 31.0f);
    float rely = (j - 1.5f) * (2.0f / 31.0f);
    float relz = (i - 1.5f) * (2.0f / 15.0f);
    float grx = (-1.0f + xi * (2.0f / 31.0f)) + relx * ax + ox;
    float gry = (-1.0f + yi * (2.0f / 31.0f)) + rely * ay + oy;
    float grz = (-1.0f + zi * (2.0f / 15.0f)) + relz * az + oz;
    float gxx = (grx + 1.0f) * 15.5f;   // (W-1)/2
    float gyy = (gry + 1.0f) * 15.5f;   // (H-1)/2
    float gzz = (grz + 1.0f) * 7.5f;    // (S-1)/2
    float fx0 = floorf(gxx), fy0 = floorf(gyy), fz0 = floorf(gzz);
    gix[tid][0] = (int)fx0; gix[tid][1] = (int)fy0; gix[tid][2] = (int)fz0;
    gfr[tid][0] = gxx - fx0; gfr[tid][1] = gyy - fy0; gfr[tid][2] = gzz - fz0;
  }
  __syncthreads();

  // --- gather Q and trilinear-sample K/V into LDS
  const float* qf = qkv + ((size_t)b * 576 + 0 * DIM + head * HD) * SHW;
  const float* kf = qkv + ((size_t)b * 576 + 1 * DIM + head * HD) * SHW;
  const float* vf = qkv + ((size_t)b * 576 + 2 * DIM + head * HD) * SHW;
  #pragma unroll
  for (int rep = 0; rep < (NWIN * HD) / 256; ++rep) {
    int idx = rep * 256 + tid;
    int t = idx >> 5;           // token
    int d = idx & 31;           // channel
    int i = t >> 4, j = (t >> 2) & 3, l = t & 3;
    int vox = (wsz * 4 + i) * 1024 + (wyh * 4 + j) * 32 + (wxw * 4 + l);
    qw[t][d] = qf[(size_t)d * SHW + vox] * SCALE_Q;

    int   x0 = gix[t][0], y0 = gix[t][1], z0 = gix[t][2];
    float fx = gfr[t][0], fy = gfr[t][1], fz = gfr[t][2];
    const float* kd = kf + (size_t)d * SHW;
    const float* vd = vf + (size_t)d * SHW;
    float ks = 0.f, vs = 0.f;
    #pragma unroll
    for (int dz = 0; dz < 2; ++dz) {
      int zz = z0 + dz; float wz = dz ? fz : (1.f - fz);
      #pragma unroll
      for (int dy = 0; dy < 2; ++dy) {
        int yy = y0 + dy; float wy = dy ? fy : (1.f - fy);
        #pragma unroll
        for (int dx = 0; dx < 2; ++dx) {
          int xx = x0 + dx; float wx = dx ? fx : (1.f - fx);
          bool ok = (zz >= 0 && zz < SS && yy >= 0 && yy < HH && xx >= 0 && xx < WW);
          float wgt = ok ? (wz * wy * wx) : 0.f;
          int zc = min(max(zz, 0), SS - 1);
          int yc = min(max(yy, 0), HH - 1);
          int xc = min(max(xx, 0), WW - 1);
          int vv = zc * 1024 + yc * 32 + xc;
          ks += kd[vv] * wgt;
          vs += vd[vv] * wgt;
        }
      }
    }
    kw[t][d] = ks;
    vw[t][d] = vs;
  }
  __syncthreads();

  // --- attn = (q*scale) . k^T : 16 tiles of 16x16, 2 per wave (unrolled)
  int lane  = tid & 31;
  int wave  = tid >> 5;
  int mrow  = lane & 15;
  int khalf = (lane >> 4) * 2;
  #pragma unroll
  for (int u = 0; u < 2; ++u) {
    int tt = wave + u * 8;
    int mt = tt >> 2, nt = tt & 3;
    v8f c = {};
    #pragma unroll
    for (int k0 = 0; k0 < HD; k0 += 4) {
      v2f a, bb;
      a.x  = qw[mt * 16 + mrow][k0 + khalf + 0];
      a.y  = qw[mt * 16 + mrow][k0 + khalf + 1];
      bb.x = kw[nt * 16 + mrow][k0 + khalf + 0];   // B[k][n] = kw[n][k]
      bb.y = kw[nt * 16 + mrow][k0 + khalf + 1];
      c = wmma_f32(a, bb, c);
    }
    int n  = nt * 16 + (lane & 15);
    int mb = mt * 16 + (lane >> 4) * 8;
    #pragma unroll
    for (int r = 0; r < 8; ++r) attn[mb + r][n] = c[r];
  }
  __syncthreads();

  // --- add relative position bias: rpi = (ds+3)*11 + (dh+3)*7 + (dw+3)
  #pragma unroll
  for (int rep = 0; rep < (NWIN * NWIN) / 256; ++rep) {
    int idx = rep * 256 + tid;
    int qt = idx >> 6, kt = idx & 63;
    int qs = qt >> 4, qh = (qt >> 2) & 3, qx = qt & 3;
    int ks2 = kt >> 4, kh = (kt >> 2) & 3, kx = kt & 3;
    int rpi = (qs - ks2 + 3) * 11 + (qh - kh + 3) * 7 + (qx - kx + 3);
    attn[qt][kt] += rpb_table[rpi * NH + head];
  }
  __syncthreads();

  // --- row softmax (64 rows, one thread each)
  if (tid < NWIN) {
    float mx = -1e30f;
    for (int kk = 0; kk < NWIN; ++kk) mx = fmaxf(mx, attn[tid][kk]);
    float sum = 0.f;
    for (int kk = 0; kk < NWIN; ++kk) {
      float e = __expf(attn[tid][kk] - mx);
      attn[tid][kk] = e;
      sum += e;
    }
    float inv = 1.0f / sum;
    for (int kk = 0; kk < NWIN; ++kk) attn[tid][kk] *= inv;
  }
  __syncthreads();

  // --- out = attn . v : 8 tiles of 16x16 (M=64, N=32, K=64), 1 per wave
  {
    int mt = wave >> 1, nt = wave & 1;
    v8f c = {};
    #pragma unroll
    for (int k0 = 0; k0 < NWIN; k0 += 4) {
      v2f a, bb;
      a.x  = attn[mt * 16 + mrow][k0 + khalf + 0];
      a.y  = attn[mt * 16 + mrow][k0 + khalf + 1];
      bb.x = vw[k0 + khalf + 0][nt * 16 + mrow];
      bb.y = vw[k0 + khalf + 1][nt * 16 + mrow];
      c = wmma_f32(a, bb, c);
    }
    int n  = nt * 16 + (lane & 15);      // channel within head
    int mb = mt * 16 + (lane >> 4) * 8;  // token base
    float* op = outp + ((size_t)b * DIM + head * HD + n) * SHW;
    #pragma unroll
    for (int r = 0; r < 8; ++r) {
      int t = mb + r;
      int vox = (wsz * 4 + (t >> 4)) * 1024
              + (wyh * 4 + ((t >> 2) & 3)) * 32
              + (wxw * 4 + (t & 3));
      op[vox] = c[r];
    }
  }
}

// ---------------------------------------------------------------------------
extern "C" void kernel_launch(void* const* d_in, const int* in_sizes, int n_in,
                              void* d_out, int out_size, void* d_ws, size_t ws_size,
                              hipStream_t stream) {
  const float* x      = (const float*)d_in[0];
  // d_in[1] (x_up) is unused by the reference
  const float* w_qkv  = (const float*)d_in[2];
  const float* b_qkv  = (const float*)d_in[3];
  const float* w_off  = (const float*)d_in[4];
  const float* b_off  = (const float*)d_in[5];
  const float* w_sc   = (const float*)d_in[6];
  const float* b_sc   = (const float*)d_in[7];
  const float* w_proj = (const float*)d_in[8];
  const float* b_proj = (const float*)d_in[9];
  const float* rpb    = (const float*)d_in[10];
  float* out = (float*)d_out;

  float* ws      = (float*)d_ws;
  float* qkv     = ws;                                   // B*576*SHW
  float* off_arr = qkv + (size_t)BB * 576 * SHW;         // B*NH*3*NW
  float* sc_arr  = off_arr + (size_t)BB * NH * 3 * NW;   // B*NH*3*NW
  float* pre     = sc_arr + (size_t)BB * NH * 3 * NW;    // B*DIM*SHW

  // 1. pooled offsets / scales
  k_pool_offsets<<<BB * NW, DIM, 0, stream>>>(x, w_off, b_off, w_sc, b_sc,
                                              off_arr, sc_arr);
  // 2. qkv projection (GEMM 576x192 x 192x16384 per batch), 64xM strips
  k_conv1x1_wmma<576><<<(BB * (576 / 64) * (SHW / 16)) / 8, 256, 0, stream>>>(
      x, w_qkv, b_qkv, qkv);
  // 3. deformable window attention
  k_attn<<<BB * NH * NW, 256, 0, stream>>>(qkv, off_arr, sc_arr, rpb, pre);
  // 4. output projection
  k_conv1x1_wmma<192><<<(BB * (192 / 64) * (SHW / 16)) / 8, 256, 0, stream>>>(
      pre, w_proj, b_proj, out);
}